// MultiHeadAttention_747324309739
// MI455X (gfx1250) — compile-verified
//
#include <hip/hip_runtime.h>
#include <math.h>

#define L 256
#define Dm 256
#define H 8
#define DKh 32
#define LDS_STRIDE 260   // 260 % 64 == 4 -> 16 rows hit 16 distinct banks

typedef float v2f __attribute__((ext_vector_type(2)));
typedef float v8f __attribute__((ext_vector_type(8)));

__device__ __forceinline__ float wave_max(float v) {
#pragma unroll
    for (int m = 16; m >= 1; m >>= 1) v = fmaxf(v, __shfl_xor(v, m, 32));
    return v;
}
__device__ __forceinline__ float wave_sum(float v) {
#pragma unroll
    for (int m = 16; m >= 1; m >>= 1) v += __shfl_xor(v, m, 32);
    return v;
}

// ---------------- LayerNorm of q (one wave per row) ----------------
__global__ __launch_bounds__(32) void ln_kernel(const float* __restrict__ x,
                                                const float* __restrict__ g,
                                                const float* __restrict__ b,
                                                float* __restrict__ y) {
    const int row = blockIdx.x;
    const int lane = threadIdx.x;
    float vals[8];
    float s = 0.f;
#pragma unroll
    for (int i = 0; i < 8; ++i) { vals[i] = x[row * Dm + lane + i * 32]; s += vals[i]; }
    const float mu = wave_sum(s) * (1.0f / Dm);
    float vv = 0.f;
#pragma unroll
    for (int i = 0; i < 8; ++i) { float d = vals[i] - mu; vv += d * d; }
    const float rs = rsqrtf(wave_sum(vv) * (1.0f / Dm) + 1e-6f);
#pragma unroll
    for (int i = 0; i < 8; ++i) {
        int c = lane + i * 32;
        y[row * Dm + c] = (vals[i] - mu) * rs * g[c] + b[c];
    }
}

// ---- C[i][n] = sum_d A[i][d] * W[n][d] + bias[n] (+ res[i][n]) ----
// one wave per 16x32 tile: one A fragment feeds two WMMAs (2 accumulators)
template <bool HAS_RES>
__global__ __launch_bounds__(256) void gemm_wmma(const float* __restrict__ A,
                                                 const float* __restrict__ W,
                                                 const float* __restrict__ bias,
                                                 const float* __restrict__ res,
                                                 float* __restrict__ C) {
    const int wave = threadIdx.x >> 5;
    const int lane = threadIdx.x & 31;
    const int tile = blockIdx.x * 8 + wave;        // 128 tiles = 16 (M) x 8 (N)
    const int tm = (tile >> 3) * 16;
    const int tn = (tile & 7) * 32;
    const int half = lane >> 4;                    // which K sub-pair
    const int l15 = lane & 15;
    const float* arow  = A + (tm + l15) * Dm + 2 * half;       // A[m][k0+2h..+1]
    const float* brow0 = W + (tn + l15) * Dm + 2 * half;       // B[k][n] = W[n][k]
    const float* brow1 = W + (tn + 16 + l15) * Dm + 2 * half;
    __builtin_prefetch(arow, 0, 3);                // global_prefetch_b8
    __builtin_prefetch(brow0, 0, 3);
    __builtin_prefetch(brow1, 0, 3);
    v8f acc0 = {}, acc1 = {};
#pragma unroll 4
    for (int k = 0; k < Dm; k += 4) {
        v2f a  = *(const v2f*)(arow + k);
        v2f b0 = *(const v2f*)(brow0 + k);
        v2f b1 = *(const v2f*)(brow1 + k);
        acc0 = __builtin_amdgcn_wmma_f32_16x16x4_f32(false, a, false, b0,
                                                     (short)0, acc0, false, false);
        acc1 = __builtin_amdgcn_wmma_f32_16x16x4_f32(false, a, false, b1,
                                                     (short)0, acc1, false, false);
    }
    const int col0 = tn + l15;
    const int col1 = tn + 16 + l15;
    const float bv0 = bias[col0];
    const float bv1 = bias[col1];
#pragma unroll
    for (int r = 0; r < 8; ++r) {
        int row = tm + r + half * 8;
        float o0 = acc0[r] + bv0;
        float o1 = acc1[r] + bv1;
        if (HAS_RES) {
            o0 += res[row * Dm + col0];
            o1 += res[row * Dm + col1];
        }
        C[row * Dm + col0] = o0;
        C[row * Dm + col1] = o1;
    }
}

// --------- s[h,j] = qbase[j,head h] . kbase[j,head h] ---------
__global__ __launch_bounds__(256) void score_kernel(const float* __restrict__ qb,
                                                    const float* __restrict__ kb,
                                                    float* __restrict__ s) {
    const int j = threadIdx.x;
#pragma unroll
    for (int h = 0; h < H; ++h) {
        float acc = 0.f;
        const float* qr = qb + j * Dm + h * DKh;
        const float* kr = kb + j * Dm + h * DKh;
#pragma unroll 8
        for (int d = 0; d < DKh; ++d) acc = fmaf(qr[d], kr[d], acc);
        s[h * L + j] = acc;
    }
}

// ---- per (head, 16-row block): logits -> softmax -> attn out,
//      then O[16x32] = Wgt[16x256] x Vhead[256x32] via WMMA ----
__global__ __launch_bounds__(256) void attn_kernel(const float* __restrict__ t,
                                                   const float* __restrict__ s,
                                                   const float* __restrict__ vb,
                                                   float* __restrict__ attn_out,
                                                   float* __restrict__ o_pre) {
    __shared__ float wgt[16 * LDS_STRIDE];
    const int iblock = blockIdx.x;   // 0..15
    const int h = blockIdx.y;        // 0..7
    const int tid = threadIdx.x;
    const float inv_temp = 0.17677669529663687f;   // 1/sqrt(32)

    // phase 1: logits[ii][j] = (1 + e^{-2|ti-tj|}) * s[h,j] / temp
    for (int idx = tid; idx < 16 * L; idx += 256) {
        int ii = idx >> 8;
        int j = idx & 255;
        int i = iblock * 16 + ii;
        float dt = fabsf(t[i] - t[j]);
        wgt[ii * LDS_STRIDE + j] = (1.0f + __expf(-2.0f * dt)) * s[h * L + j] * inv_temp;
    }
    __syncthreads();

    // phase 2: softmax per row (wave w owns rows 2w, 2w+1); write attn,
    // rebuild LDS as post-softmax weights * (1 + e^{-dt})
    const int wave = tid >> 5, lane = tid & 31;
#pragma unroll
    for (int rr = 0; rr < 2; ++rr) {
        int ii = wave * 2 + rr;
        int i = iblock * 16 + ii;
        float ti = t[i];
        float lv[8];
        float m = -1e30f;
#pragma unroll
        for (int c = 0; c < 8; ++c) {
            lv[c] = wgt[ii * LDS_STRIDE + lane + c * 32];
            m = fmaxf(m, lv[c]);
        }
        m = wave_max(m);
        float sum = 0.f;
#pragma unroll
        for (int c = 0; c < 8; ++c) { lv[c] = __expf(lv[c] - m); sum += lv[c]; }
        const float inv = 1.0f / wave_sum(sum);
#pragma unroll
        for (int c = 0; c < 8; ++c) {
            int j = lane + c * 32;
            float a = lv[c] * inv;
            attn_out[(h * L + i) * L + j] = a;
            float dt = fabsf(ti - t[j]);
            wgt[ii * LDS_STRIDE + j] = a * (1.0f + __expf(-dt));
        }
    }
    __syncthreads();

    // phase 3: waves 0,1 each compute one 16x16 output tile of O[16x32]
    if (wave < 2) {
        const int ntile = wave;
        const int half = lane >> 4, l15 = lane & 15;
        const float* bbase = vb + h * DKh + ntile * 16 + l15;  // B[k][n]=vb[k][col]
        v8f acc = {};
#pragma unroll 4
        for (int k = 0; k < L; k += 4) {
            int ak = k + 2 * half;
            v2f a, b;
            a.x = wgt[l15 * LDS_STRIDE + ak];
            a.y = wgt[l15 * LDS_STRIDE + ak + 1];
            b.x = bbase[ak * Dm];
            b.y = bbase[(ak + 1) * Dm];
            acc = __builtin_amdgcn_wmma_f32_16x16x4_f32(false, a, false, b,
                                                        (short)0, acc, false, false);
        }
#pragma unroll
        for (int r = 0; r < 8; ++r) {
            int i = iblock * 16 + r + half * 8;
            o_pre[i * Dm + h * DKh + ntile * 16 + l15] = acc[r];
        }
    }
}

extern "C" void kernel_launch(void* const* d_in, const int* in_sizes, int n_in,
                              void* d_out, int out_size, void* d_ws, size_t ws_size,
                              hipStream_t stream) {
    const float* q    = (const float*)d_in[0];
    const float* k    = (const float*)d_in[1];
    const float* v    = (const float*)d_in[2];
    const float* t    = (const float*)d_in[3];
    const float* Wq   = (const float*)d_in[4];
    const float* bq   = (const float*)d_in[5];
    const float* Wk   = (const float*)d_in[6];
    const float* bk   = (const float*)d_in[7];
    const float* Wv   = (const float*)d_in[8];
    const float* bv   = (const float*)d_in[9];
    const float* fc_w = (const float*)d_in[10];
    const float* fc_b = (const float*)d_in[11];
    const float* ln_g = (const float*)d_in[12];
    const float* ln_b = (const float*)d_in[13];

    float* out  = (float*)d_out;          // [L, D]    (65536)
    float* attn = out + L * Dm;           // [H, L, L] (524288)

    float* ws = (float*)d_ws;
    float* qn = ws;                 // 65536
    float* qb = ws + 1 * 65536;     // 65536
    float* kb = ws + 2 * 65536;     // 65536
    float* vb = ws + 3 * 65536;     // 65536
    float* op = ws + 4 * 65536;     // 65536
    float* sc = ws + 5 * 65536;     // 2048

    ln_kernel<<<L, 32, 0, stream>>>(q, ln_g, ln_b, qn);
    gemm_wmma<false><<<16, 256, 0, stream>>>(qn, Wq, bq, nullptr, qb);
    gemm_wmma<false><<<16, 256, 0, stream>>>(k,  Wk, bk, nullptr, kb);
    gemm_wmma<false><<<16, 256, 0, stream>>>(v,  Wv, bv, nullptr, vb);
    score_kernel<<<1, 256, 0, stream>>>(qb, kb, sc);
    dim3 ag(16, H);
    attn_kernel<<<ag, 256, 0, stream>>>(t, sc, vb, attn, op);
    gemm_wmma<true><<<16, 256, 0, stream>>>(op, fc_w, fc_b, q, out);
}